// LatticeCompanderBlock2BananaSplit_70274254897549
// MI455X (gfx1250) — compile-verified
//
#include <hip/hip_runtime.h>

typedef float v2f __attribute__((ext_vector_type(2)));
typedef float v8f __attribute__((ext_vector_type(8)));

#define SQRT3F      1.7320508075688772f
#define HSQRT3F     0.8660254037844386f
#define VOR_VOL_F   0.177978515625f     /* (sqrt(3)/2)^12 == (3/4)^6 exactly */

#define KP        32                    /* K-panel size (floats)            */
#define SH_STRIDE 36                    /* padded LDS row stride: 144B =    */
                                        /* 16B-aligned for b128, bank-split */

__device__ __forceinline__ float fast_tanh(float x) {
    // tanh(x) = 1 - 2/(exp(2x)+1); saturates correctly at +/-inf, no NaN.
    float e = __expf(2.0f * x);
    return 1.0f - 2.0f / (e + 1.0f);
}

__device__ __forceinline__ float softplusf(float x) {
    return fmaxf(x, 0.0f) + log1pf(__expf(-fabsf(x)));
}

__device__ __forceinline__ void hexq(float p0, float p1, float* qx, float* qy) {
    // coset 0: rectangular lattice (a, b*sqrt3)
    float a0 = rintf(p0);
    float a1 = rintf(p1 / SQRT3F) * SQRT3F;
    // coset 1: shifted by (0.5, sqrt3/2)
    float b0 = rintf(p0 - 0.5f) + 0.5f;
    float b1 = rintf((p1 - HSQRT3F) / SQRT3F) * SQRT3F + HSQRT3F;
    float d0 = (p0 - a0) * (p0 - a0) + (p1 - a1) * (p1 - a1);
    float d1 = (p0 - b0) * (p0 - b0) + (p1 - b1) * (p1 - b1);
    *qx = (d0 <= d1) ? a0 : b0;
    *qy = (d0 <= d1) ? a1 : b1;
}

// async 16B memory->LDS copy (ASYNCcnt-tracked); ldsOff = LDS byte offset,
// gaddr = 64-bit global address. Per-lane transfer under EXEC.
__device__ __forceinline__ void async_copy_b128(unsigned ldsOff, const float* g) {
    asm volatile("global_load_async_to_lds_b128 %0, %1, off"
                 :: "v"(ldsOff), "v"((unsigned long long)(size_t)g)
                 : "memory");
}
__device__ __forceinline__ void wait_async0() {
    asm volatile("s_wait_asynccnt 0" ::: "memory");
}

// ---------------------------------------------------------------------------
// GEMM: Y[B,O] = act(X[B,I] @ W[O,I]^T + bias[O]),  leaky_relu(0.01) if leaky
// Block = 128 threads (4 waves) -> 64x64 output tile. K-panels of 32 are
// staged into LDS with GLOBAL_LOAD_ASYNC_TO_LDS_B128 (X panel 64x32, W panel
// 64x32 shared by all 4 waves). OOB W rows are zero-filled at staging time
// (ds_store_b128) so the K loop carries no masks: each wave's 16(M)x64(N)
// strip is one A fragment (ds_load_b64) feeding 4x V_WMMA_F32_16X16X4_F32.
// Fragment layouts (f32):
//   A (16x4 MxK): lane (l=lane&15, hi=lane>>4) holds M=l, VGPR v <-> K=v+2*hi
//   B (4x16 KxN): lane holds N=l, VGPR v <-> K=v+2*hi
//   C/D:          VGPR r <-> M=r+8*hi, N=l
// Requires I%4==0, B%64==0 (true: I in {24,500}, B=2048). O guarded.
// ---------------------------------------------------------------------------
__global__ __launch_bounds__(128) void gemm_wmma_tiled(
    const float* __restrict__ X, const float* __restrict__ W,
    const float* __restrict__ bias, float* __restrict__ Y,
    int I, int O, int leaky) {
    __shared__ float shX[64][SH_STRIDE];
    __shared__ float shW[64][SH_STRIDE];

    int tid  = threadIdx.x;
    int lane = tid & 31;
    int wv   = tid >> 5;        // wave id 0..3 -> M subtile
    int hi   = lane >> 4;
    int l    = lane & 15;
    int m0   = blockIdx.y * 64;
    int o0   = blockIdx.x * 64;

    unsigned ldsX = (unsigned)(size_t)&shX[0][0];   // addr[31:0] == LDS offset
    unsigned ldsW = (unsigned)(size_t)&shW[0][0];

    v8f acc[4] = {};

    for (int k0 = 0; k0 < I; k0 += KP) {
        int kw    = (I - k0 < KP) ? (I - k0) : KP;  // 32, 20 (I=500) or 24
        int cpr   = kw >> 2;                        // 16B chunks per row
        int total = 64 * cpr;

        __syncthreads();                            // LDS free to overwrite
        for (int id = tid; id < total; id += 128) {
            int r  = id / cpr;
            int ck = id - r * cpr;
            unsigned doff = (unsigned)(r * SH_STRIDE + ck * 4) * 4u;
            // X panel: rows m0..m0+63 (always in range)
            async_copy_b128(ldsX + doff, X + (size_t)(m0 + r) * I + k0 + ck * 4);
            // W panel: rows o0..o0+63; OOB rows -> zero-fill (no masks in K loop)
            int o = o0 + r;
            if (o < O) {
                async_copy_b128(ldsW + doff, W + (size_t)o * I + k0 + ck * 4);
            } else {
                float4 z = {0.0f, 0.0f, 0.0f, 0.0f};
                *(float4*)&shW[r][ck * 4] = z;      // 16B-aligned ds_store_b128
            }
        }
        wait_async0();
        __syncthreads();                            // panels visible to all

        for (int kk = 0; kk < kw; kk += 4) {
            float2 av = *(const float2*)&shX[wv * 16 + l][kk + 2 * hi];
            v2f a; a[0] = av.x; a[1] = av.y;
#pragma unroll
            for (int n = 0; n < 4; ++n) {
                float2 bv = *(const float2*)&shW[n * 16 + l][kk + 2 * hi];
                v2f b; b[0] = bv.x; b[1] = bv.y;
                acc[n] = __builtin_amdgcn_wmma_f32_16x16x4_f32(
                             false, a, false, b, (short)0, acc[n], false, false);
            }
        }
    }

#pragma unroll
    for (int n = 0; n < 4; ++n) {
        int col = o0 + n * 16 + l;
        if (col < O) {
            float bb = bias[col];
#pragma unroll
            for (int r = 0; r < 8; ++r) {
                float v = acc[n][r] + bb;
                if (leaky) v = (v > 0.0f) ? v : 0.01f * v;
                Y[(size_t)(m0 + wv * 16 + r + 8 * hi) * O + col] = v;
            }
        }
    }
}

// ---------------------------------------------------------------------------
// y_q = hex_quantize(y1), per 2-D pair. n = B * 12 pairs.
// ---------------------------------------------------------------------------
__global__ void quantize_y(const float* __restrict__ y1,
                           float* __restrict__ yq, int n) {
    int i = blockIdx.x * blockDim.x + threadIdx.x;
    if (i >= n) return;
    float p0 = y1[2 * i], p1 = y1[2 * i + 1], qx, qy;
    hexq(p0, p1, &qx, &qy);
    yq[2 * i]     = qx;
    yq[2 * i + 1] = qy;
}

// ---------------------------------------------------------------------------
// u2 = (u reshaped pairs) @ G2; u2 -= hex_quantize(u2). Stored TRANSPOSED:
// u2t[c*256 + s] for coalesced reads in the density kernel. 256*12 pairs.
// ---------------------------------------------------------------------------
__global__ void prep_u(const float* __restrict__ u, float* __restrict__ u2t) {
    int i = blockIdx.x * blockDim.x + threadIdx.x;
    if (i >= 256 * 12) return;
    int s = i / 12, p = i % 12;
    float a = u[s * 24 + 2 * p], b = u[s * 24 + 2 * p + 1];
    float v0 = a + 0.5f * b;        // row-vector @ G2
    float v1 = HSQRT3F * b;
    float qx, qy;
    hexq(v0, v1, &qx, &qy);
    u2t[(2 * p) * 256 + s]     = v0 - qx;
    u2t[(2 * p + 1) * 256 + s] = v1 - qy;
}

// ---------------------------------------------------------------------------
// Precompute per-channel density params (hoisted: shared by all 12.6M evals).
// Packed layout per channel c (stride 44):
//  [0..2] W0=softplus(m0)  [3..11] W1  [12..20] W2  [21..23] W3
//  [24..26] b0  [27..29] b1  [30..32] b2  [33] b3
//  [34..36] tanh(f0)  [37..39] tanh(f1)  [40..42] tanh(f2)  [43] pad
// ---------------------------------------------------------------------------
__global__ void prep_params(
    const float* __restrict__ m0, const float* __restrict__ m1,
    const float* __restrict__ m2, const float* __restrict__ m3,
    const float* __restrict__ b0, const float* __restrict__ b1,
    const float* __restrict__ b2, const float* __restrict__ b3,
    const float* __restrict__ f0, const float* __restrict__ f1,
    const float* __restrict__ f2, float* __restrict__ params) {
    int c = threadIdx.x;
    if (c >= 24) return;
    float* p = params + c * 44;
#pragma unroll
    for (int f = 0; f < 3; ++f) p[f]      = softplusf(m0[c * 3 + f]);
#pragma unroll
    for (int k = 0; k < 9; ++k) p[3 + k]  = softplusf(m1[c * 9 + k]);
#pragma unroll
    for (int k = 0; k < 9; ++k) p[12 + k] = softplusf(m2[c * 9 + k]);
#pragma unroll
    for (int f = 0; f < 3; ++f) p[21 + f] = softplusf(m3[c * 3 + f]);
#pragma unroll
    for (int f = 0; f < 3; ++f) p[24 + f] = b0[c * 3 + f];
#pragma unroll
    for (int f = 0; f < 3; ++f) p[27 + f] = b1[c * 3 + f];
#pragma unroll
    for (int f = 0; f < 3; ++f) p[30 + f] = b2[c * 3 + f];
    p[33] = b3[c];
#pragma unroll
    for (int f = 0; f < 3; ++f) p[34 + f] = fast_tanh(f0[c * 3 + f]);
#pragma unroll
    for (int f = 0; f < 3; ++f) p[37 + f] = fast_tanh(f1[c * 3 + f]);
#pragma unroll
    for (int f = 0; f < 3; ++f) p[40 + f] = fast_tanh(f2[c * 3 + f]);
    p[43] = 0.0f;
}

// ---------------------------------------------------------------------------
// lik[b] = vor_vol * (1/256) * sum_s prod_c density(yq[b,c] + u2[s,c], c)
// One block per batch element, one thread per MC sample.
// ---------------------------------------------------------------------------
__global__ __launch_bounds__(256) void density_lik(
    const float* __restrict__ yq, const float* __restrict__ u2t,
    const float* __restrict__ params, float* __restrict__ lik) {
    __shared__ float P[24 * 44];
    __shared__ float yqs[24];
    __shared__ float red[256];
    int b = blockIdx.x;
    int t = threadIdx.x;
    for (int i = t; i < 24 * 44; i += 256) P[i] = params[i];
    if (t < 24) yqs[t] = yq[b * 24 + t];
    __syncthreads();

    float pr = 1.0f;
#pragma unroll 1
    for (int c = 0; c < 24; ++c) {
        const float* p = &P[c * 44];
        float xv = yqs[c] + u2t[c * 256 + t];
        float h0[3], J0[3];
#pragma unroll
        for (int f = 0; f < 3; ++f) {             // layer 0 (fi = 1)
            float a  = p[f] * xv + p[24 + f];
            float th = fast_tanh(a);
            float T  = p[34 + f];
            h0[f] = a + T * th;
            J0[f] = (1.0f + T * (1.0f - th * th)) * p[f];
        }
        float h1[3], J1[3];
#pragma unroll
        for (int i = 0; i < 3; ++i) {             // layer 1
            const float* w = &p[3 + i * 3];
            float a  = w[0] * h0[0] + w[1] * h0[1] + w[2] * h0[2] + p[27 + i];
            float wj = w[0] * J0[0] + w[1] * J0[1] + w[2] * J0[2];
            float th = fast_tanh(a);
            float T  = p[37 + i];
            h1[i] = a + T * th;
            J1[i] = (1.0f + T * (1.0f - th * th)) * wj;
        }
        float h2[3], J2[3];
#pragma unroll
        for (int i = 0; i < 3; ++i) {             // layer 2
            const float* w = &p[12 + i * 3];
            float a  = w[0] * h1[0] + w[1] * h1[1] + w[2] * h1[2] + p[30 + i];
            float wj = w[0] * J1[0] + w[1] * J1[1] + w[2] * J1[2];
            float th = fast_tanh(a);
            float T  = p[40 + i];
            h2[i] = a + T * th;
            J2[i] = (1.0f + T * (1.0f - th * th)) * wj;
        }
        float L  = p[21] * h2[0] + p[22] * h2[1] + p[23] * h2[2] + p[33];
        float JL = p[21] * J2[0] + p[22] * J2[1] + p[23] * J2[2];
        float e  = __expf(-L);
        float sg = 1.0f / (1.0f + e);
        pr *= sg * (1.0f - sg) * JL;
    }

    red[t] = pr;
    __syncthreads();
    for (int off = 128; off > 0; off >>= 1) {
        if (t < off) red[t] += red[t + off];
        __syncthreads();
    }
    if (t == 0) lik[b] = VOR_VOL_F * red[0] * (1.0f / 256.0f);
}

// ---------------------------------------------------------------------------
extern "C" void kernel_launch(void* const* d_in, const int* in_sizes, int n_in,
                              void* d_out, int out_size, void* d_ws, size_t ws_size,
                              hipStream_t stream) {
    const int B = 2048, N = 24, DH = 500;

    const float* x = (const float*)d_in[0];
    const float* u = (const float*)d_in[1];
    const float* ga_w[4] = {(const float*)d_in[2], (const float*)d_in[4],
                            (const float*)d_in[6], (const float*)d_in[8]};
    const float* ga_b[4] = {(const float*)d_in[3], (const float*)d_in[5],
                            (const float*)d_in[7], (const float*)d_in[9]};
    const float* gs_w[4] = {(const float*)d_in[10], (const float*)d_in[12],
                            (const float*)d_in[14], (const float*)d_in[16]};
    const float* gs_b[4] = {(const float*)d_in[11], (const float*)d_in[13],
                            (const float*)d_in[15], (const float*)d_in[17]};
    const float* eb_m[4] = {(const float*)d_in[18], (const float*)d_in[19],
                            (const float*)d_in[20], (const float*)d_in[21]};
    const float* eb_b[4] = {(const float*)d_in[22], (const float*)d_in[23],
                            (const float*)d_in[24], (const float*)d_in[25]};
    const float* eb_f[3] = {(const float*)d_in[26], (const float*)d_in[27],
                            (const float*)d_in[28]};

    float* ws     = (float*)d_ws;
    float* bufA   = ws;                       // 2048*512 floats
    float* bufB   = bufA + (size_t)B * 512;   // 2048*512 floats
    float* yq     = bufB + (size_t)B * 512;   // 2048*24
    float* u2t    = yq + (size_t)B * N;       // 24*256 (transposed)
    float* params = u2t + 24 * 256;           // 24*44
    float* xhat   = (float*)d_out;            // 2048*24
    float* lik    = xhat + (size_t)B * N;     // 2048

    dim3 blk(128);
    auto gemm = [&](const float* X, const float* W, const float* bias,
                    float* Y, int I, int O, int leaky) {
        dim3 grid((O + 63) / 64, B / 64);
        gemm_wmma_tiled<<<grid, blk, 0, stream>>>(X, W, bias, Y, I, O, leaky);
    };

    // analysis transform: x -> y1 (bufB)
    gemm(x,    ga_w[0], ga_b[0], bufA, N,  DH, 1);
    gemm(bufA, ga_w[1], ga_b[1], bufB, DH, DH, 1);
    gemm(bufB, ga_w[2], ga_b[2], bufA, DH, DH, 1);
    gemm(bufA, ga_w[3], ga_b[3], bufB, DH, N,  0);

    // lattice quantize + MC-sample prep + density-param hoisting
    quantize_y<<<(B * 12 + 255) / 256, 256, 0, stream>>>(bufB, yq, B * 12);
    prep_u<<<(256 * 12 + 255) / 256, 256, 0, stream>>>(u, u2t);
    prep_params<<<1, 32, 0, stream>>>(eb_m[0], eb_m[1], eb_m[2], eb_m[3],
                                      eb_b[0], eb_b[1], eb_b[2], eb_b[3],
                                      eb_f[0], eb_f[1], eb_f[2], params);

    // synthesis transform: yq -> x_hat (straight into d_out)
    gemm(yq,   gs_w[0], gs_b[0], bufA, N,  DH, 1);
    gemm(bufA, gs_w[1], gs_b[1], bufB, DH, DH, 1);
    gemm(bufB, gs_w[2], gs_b[2], bufA, DH, DH, 1);
    gemm(bufA, gs_w[3], gs_b[3], xhat, DH, N,  0);

    // MC Voronoi likelihood
    density_lik<<<B, 256, 0, stream>>>(yq, u2t, params, lik);
}